// InterpretableBCosGATLayer_28346784153669
// MI455X (gfx1250) — compile-verified
//
#include <hip/hip_runtime.h>
#include <cstddef>

// ---------------------------------------------------------------------------
// InterpretableBCosGAT layer for MI455X (gfx1250, wave32).
//
// Pipeline:
//   K1 gemm_wmma<128> : h = x @ lin_w^T              (v_wmma_f32_16x16x4_f32)
//   K2 head_invnorm   : inv_n[node,head] = 1/||h_head||
//   K3 zero_f32       : out_accum = 0
//   K4 edge_scatter   : cos-scaled messages, global_atomic_add_f32 scatter
//   K5 w_invnorm      : inv_wn[j] = 1/||bcos_w row j||
//   K6 gemm_wmma<64>  : lin = out_accum @ bcos_w^T   -> written into d_out
//   K7 epilogue       : bcos rescale (cos2 = lin * inv_on * inv_wn) + LayerNorm
//
// GEMM design: one wave owns 4 row-tiles x 4 col-tiles (64x64 output block).
// Per K-step (K+=4): 8 global_load_b64 feed 16 independent v_wmma chains, so
// load latency is amortized 16x (vs 1x in the naive wave-per-16-rows form)
// and the 16 independent accumulators give the scheduler ILP across waits.
// ---------------------------------------------------------------------------

typedef __attribute__((ext_vector_type(2))) float v2f;
typedef __attribute__((ext_vector_type(8))) float v8f;

#define NNODES 50000
#define IN_CH  128
#define HID    64
#define NHEADS 4
#define DHEAD  16
#define NEDGES 800000
#define EPS_C  1e-6f
#define LN_EPS 1e-5f

// ---------------------------------------------------------------------------
// K1/K6: C[M,N] = A[M,K] @ W[N,K]^T  via V_WMMA_F32_16X16X4_F32, wave32.
// A fragment: lane L -> row (L&15), float2 at K = k0 + ((L>>4)<<1).
// B fragment: lane L -> col (L&15), float2 at W[col*K + k0 + koff].
// D store:    VGPR v -> row v + ((L>>4)<<3), col (L&15) + 16*t.
// 4 row tiles per wave; A-row reads clamped, stores predicated for the tail
// (WMMA itself always runs with EXEC all-ones).
// ---------------------------------------------------------------------------
template <int K>
__global__ __launch_bounds__(32)
void gemm_wmma(const float* __restrict__ A, const float* __restrict__ W,
               float* __restrict__ C, int nrows) {
  const int lane  = threadIdx.x;          // 0..31 (wave32)
  const int mcol  = lane & 15;
  const int koff  = (lane >> 4) << 1;     // 0 or 2
  const int tile0 = blockIdx.x * 4;       // first of 4 row-tiles

  const float* a[4];
#pragma unroll
  for (int i = 0; i < 4; ++i) {
    int row = (tile0 + i) * 16 + mcol;
    row = row < nrows ? row : nrows - 1;  // clamp: valid address, maybe garbage
    a[i] = A + (size_t)row * K + koff;
  }
  const float* w[4];
#pragma unroll
  for (int t = 0; t < 4; ++t)
    w[t] = W + (size_t)(t * 16 + mcol) * K + koff;

  v8f acc[4][4];
#pragma unroll
  for (int i = 0; i < 4; ++i)
#pragma unroll
    for (int t = 0; t < 4; ++t) acc[i][t] = (v8f){};

  for (int k0 = 0; k0 < K; k0 += 4) {
    v2f bf[4], af[4];
#pragma unroll
    for (int t = 0; t < 4; ++t) bf[t] = *(const v2f*)(w[t] + k0);
#pragma unroll
    for (int i = 0; i < 4; ++i) af[i] = *(const v2f*)(a[i] + k0);
#pragma unroll
    for (int i = 0; i < 4; ++i)
#pragma unroll
      for (int t = 0; t < 4; ++t)
        acc[i][t] = __builtin_amdgcn_wmma_f32_16x16x4_f32(
            false, af[i], false, bf[t], (short)0, acc[i][t], false, false);
  }

#pragma unroll
  for (int i = 0; i < 4; ++i) {
    const int mbase = (tile0 + i) * 16 + ((lane >> 4) << 3);  // +0 or +8
#pragma unroll
    for (int v = 0; v < 8; ++v) {
      const int row = mbase + v;
      if (row < nrows) {
        float* r = C + (size_t)row * HID + mcol;
        r[0]  = acc[i][0][v];
        r[16] = acc[i][1][v];
        r[32] = acc[i][2][v];
        r[48] = acc[i][3][v];
      }
    }
  }
}

// ---------------------------------------------------------------------------
// K2: inverse L2 norm per (node, head) sub-vector of 16.
// h layout is [N, HEADS*DHEAD] row-major => flat (node*4 + head)*16.
// ---------------------------------------------------------------------------
__global__ __launch_bounds__(256)
void head_invnorm(const float* __restrict__ H, float* __restrict__ inv_n, int total) {
  int t = blockIdx.x * blockDim.x + threadIdx.x;
  if (t >= total) return;
  const float4* p = (const float4*)(H + (size_t)t * DHEAD);
  float s = 0.f;
#pragma unroll
  for (int i = 0; i < 4; ++i) {
    float4 q = p[i];
    s += q.x * q.x + q.y * q.y + q.z * q.z + q.w * q.w;
  }
  inv_n[t] = 1.0f / fmaxf(sqrtf(s), 1e-12f);   // matches F.normalize eps
}

__global__ __launch_bounds__(256)
void zero_f32(float* __restrict__ p, int n) {
  int t = blockIdx.x * blockDim.x + threadIdx.x;
  if (t < n) p[t] = 0.0f;
}

// ---------------------------------------------------------------------------
// K4: one wave per edge. Lane l owns channels 2l, 2l+1; head = l>>3.
// cos = (sum_{d} h_src*h_dst) * inv_n_src * inv_n_dst, clipped to [eps,1].
// B_EXP=2.0 => scale = cos^(B-1) = cos exactly. Scatter via f32 HW atomics.
// ---------------------------------------------------------------------------
__global__ __launch_bounds__(256)
void edge_scatter(const int* __restrict__ erow, const int* __restrict__ ecol,
                  const float* __restrict__ H, const float* __restrict__ inv_n,
                  float* __restrict__ out_accum, int E) {
  int e = blockIdx.x * 8 + (threadIdx.x >> 5);
  if (e >= E) return;                       // uniform per wave
  const int lane = threadIdx.x & 31;
  const int ch   = lane << 1;
  const int head = lane >> 3;

  const int dst = erow[e];                  // target (row)
  const int src = ecol[e];                  // source (col)

  v2f hs = *(const v2f*)(H + (size_t)src * HID + ch);
  v2f hd = *(const v2f*)(H + (size_t)dst * HID + ch);

  float p = hs.x * hd.x + hs.y * hd.y;      // per-lane partial dot
  p += __shfl_xor(p, 1, 32);                // reduce across the 8 lanes of one head
  p += __shfl_xor(p, 2, 32);
  p += __shfl_xor(p, 4, 32);

  float c = p * inv_n[src * NHEADS + head] * inv_n[dst * NHEADS + head];
  c = fminf(fmaxf(c, EPS_C), 1.0f);         // clip; scale == c since B_EXP-1 == 1

  float* o = out_accum + (size_t)dst * HID + ch;
  unsafeAtomicAdd(o,     hs.x * c);         // global_atomic_add_f32
  unsafeAtomicAdd(o + 1, hs.y * c);
}

// ---------------------------------------------------------------------------
// K5: inverse row norms of bcos_w [64,64]. One block of 64 threads.
// ---------------------------------------------------------------------------
__global__ __launch_bounds__(64)
void w_invnorm(const float* __restrict__ W, float* __restrict__ inv_wn) {
  int j = threadIdx.x;
  const float* r = W + (size_t)j * HID;
  float s = 0.f;
#pragma unroll
  for (int i = 0; i < HID; ++i) s += r[i] * r[i];
  inv_wn[j] = 1.0f / fmaxf(sqrtf(s), 1e-12f);
}

// ---------------------------------------------------------------------------
// K7: per-node B-cos rescale + LayerNorm. One wave per node; lane l owns
// channels 2l, 2l+1. `out` already holds lin = out_accum @ bcos_w^T; each
// wave reads its own row then overwrites it (no cross-wave hazards).
// ---------------------------------------------------------------------------
__global__ __launch_bounds__(256)
void epilogue(const float* __restrict__ out_accum, const float* __restrict__ inv_wn,
              const float* __restrict__ gamma, const float* __restrict__ beta,
              float* __restrict__ out, int N) {
  int node = blockIdx.x * 8 + (threadIdx.x >> 5);
  if (node >= N) return;                    // uniform per wave
  const int lane = threadIdx.x & 31;
  const int ch   = lane << 1;

  v2f o = *(const v2f*)(out_accum + (size_t)node * HID + ch);
  v2f l = *(const v2f*)(out       + (size_t)node * HID + ch);

  // ||out row||
  float ss = o.x * o.x + o.y * o.y;
#pragma unroll
  for (int m = 1; m < 32; m <<= 1) ss += __shfl_xor(ss, m, 32);
  const float inv_on = 1.0f / fmaxf(sqrtf(ss), 1e-12f);

  // cos2 = lin * inv_on * inv_wn, clipped; out_bcos = lin * cos2
  float c0 = fminf(fmaxf(l.x * inv_on * inv_wn[ch],     EPS_C), 1.0f);
  float c1 = fminf(fmaxf(l.y * inv_on * inv_wn[ch + 1], EPS_C), 1.0f);
  float b0 = l.x * c0;
  float b1 = l.y * c1;

  // LayerNorm over 64 channels
  float sum = b0 + b1;
#pragma unroll
  for (int m = 1; m < 32; m <<= 1) sum += __shfl_xor(sum, m, 32);
  const float mu = sum * (1.0f / HID);
  float d0 = b0 - mu, d1 = b1 - mu;
  float vs = d0 * d0 + d1 * d1;
#pragma unroll
  for (int m = 1; m < 32; m <<= 1) vs += __shfl_xor(vs, m, 32);
  const float rstd = rsqrtf(vs * (1.0f / HID) + LN_EPS);

  float* w = out + (size_t)node * HID + ch;
  w[0] = d0 * rstd * gamma[ch]     + beta[ch];
  w[1] = d1 * rstd * gamma[ch + 1] + beta[ch + 1];
}

// ---------------------------------------------------------------------------
extern "C" void kernel_launch(void* const* d_in, const int* in_sizes, int n_in,
                              void* d_out, int out_size, void* d_ws, size_t ws_size,
                              hipStream_t stream) {
  const float* x      = (const float*)d_in[0];
  const int*   eidx   = (const int*)d_in[1];     // [2,E] flat: row then col
  const float* lin_w  = (const float*)d_in[2];   // [64,128]
  const float* bcos_w = (const float*)d_in[3];   // [64,64]
  const float* gamma  = (const float*)d_in[4];
  const float* beta   = (const float*)d_in[5];
  float*       out    = (float*)d_out;

  // workspace layout (floats): h | out_accum | inv_n | inv_wn
  float* h         = (float*)d_ws;
  float* out_accum = h + (size_t)NNODES * HID;
  float* inv_n     = out_accum + (size_t)NNODES * HID;
  float* inv_wn    = inv_n + (size_t)NNODES * NHEADS;

  const int* erow = eidx;
  const int* ecol = eidx + NEDGES;

  const int ntiles = (NNODES + 15) / 16;             // 3125
  const int gblocks = (ntiles + 3) / 4;              // 782 waves, 64 rows each

  // K1: h = x @ lin_w^T
  gemm_wmma<IN_CH><<<gblocks, 32, 0, stream>>>(x, lin_w, h, NNODES);

  // K2: per-(node,head) inverse norms
  {
    int total = NNODES * NHEADS;
    head_invnorm<<<(total + 255) / 256, 256, 0, stream>>>(h, inv_n, total);
  }

  // K3: zero the scatter accumulator
  {
    int total = NNODES * HID;
    zero_f32<<<(total + 255) / 256, 256, 0, stream>>>(out_accum, total);
  }

  // K4: edge messages + scatter-add (dominant cost: ~0.6 GB traffic + atomics)
  edge_scatter<<<(NEDGES + 7) / 8, 256, 0, stream>>>(erow, ecol, h, inv_n,
                                                     out_accum, NEDGES);

  // K5: bcos_w row inverse norms
  w_invnorm<<<1, 64, 0, stream>>>(bcos_w, inv_wn);

  // K6: lin = out_accum @ bcos_w^T, written straight into d_out
  gemm_wmma<HID><<<gblocks, 32, 0, stream>>>(out_accum, bcos_w, out, NNODES);

  // K7: bcos rescale + LayerNorm, in place on d_out
  epilogue<<<(NNODES + 7) / 8, 256, 0, stream>>>(out_accum, inv_wn, gamma, beta,
                                                 out, NNODES);
}